// RCNN_45552423141651
// MI455X (gfx1250) — compile-verified
//
#include <hip/hip_runtime.h>
#include <math.h>

// Problem constants from the reference
#define ROI_C   512
#define ROI_H   14
#define ROI_W   14
#define ROI_P   7
#define CHUNK   64                        // channels staged per LDS chunk
#define NCHUNK  (ROI_C / CHUNK)           // 8
#define MAXROWS 8                         // max hspan (proved <= 8)
#define CH_F    (MAXROWS * ROI_W)         // 112 floats of LDS per channel
#define TILE_F  (CHUNK * CH_F + 16)       // +pad: unconditional 2x2 reads stay in-bounds
#define TPB     256
#define NQ      ((CHUNK * 49) / 4)        // 784 float4 outputs per chunk

#ifndef __has_builtin
#define __has_builtin(x) 0
#endif

#if __has_builtin(__builtin_amdgcn_global_load_async_to_lds_b64)
#define HAVE_ASYNC_B64 1
#else
#define HAVE_ASYNC_B64 0
#endif

typedef int v2i __attribute__((ext_vector_type(2)));
typedef __attribute__((address_space(1))) v2i* gv2i_ptr;   // global
typedef __attribute__((address_space(3))) v2i* lv2i_ptr;   // LDS

template <int N>
__device__ __forceinline__ void wait_async() {
#if __has_builtin(__builtin_amdgcn_s_wait_asynccnt)
  __builtin_amdgcn_s_wait_asynccnt(N);
#else
  asm volatile("s_wait_asynccnt %0" :: "n"(N) : "memory");
#endif
}

#if HAVE_ASYNC_B64
// Exactly 14 async b64 loads per thread (uniform per-wave ASYNCcnt batch of 14;
// out-of-span slots re-copy the last word -> benign duplicate LDS writes).
__device__ __forceinline__ void stage_chunk_async(const float* gchan, float* lchan,
                                                  int spanq, int t4) {
  #pragma unroll
  for (int k = 0; k < 14; ++k) {
    int r  = t4 + 4 * k;
    int rq = (r < spanq) ? r : (spanq - 1);
    __builtin_amdgcn_global_load_async_to_lds_b64(
        (gv2i_ptr)(gchan + 2 * rq), (lv2i_ptr)(lchan + 2 * rq), 0, 0);
  }
}
#endif

// Branch-free adaptive max-pool: window per axis is provably 1 or 2 wide.
// Each thread produces 4 consecutive outputs -> one b128 store.
__device__ __forceinline__ void compute_chunk(const float* __restrict__ buf,
                                              const int* __restrict__ s_win,
                                              float* __restrict__ outp) {
  for (int iq = threadIdx.x; iq < NQ; iq += TPB) {
    const int base = iq * 4;
    float v[4];
    #pragma unroll
    for (int j = 0; j < 4; ++j) {
      const unsigned idx = (unsigned)(base + j);
      const unsigned c   = idx / 49u;
      const unsigned p   = idx - c * 49u;
      const int w = s_win[p];
      const float* q = buf + c * CH_F + (w & 0xFF);
      const float a = q[0], b = q[1], e = q[ROI_W], f = q[ROI_W + 1];
      const float r0 = (w & 0x100) ? fmaxf(a, b) : a;   // row 0 max
      const float r1 = (w & 0x100) ? fmaxf(e, f) : e;   // row 1 max (masked below)
      v[j] = (w & 0x200) ? fmaxf(r0, r1) : r0;
    }
    *(float4*)(outp + base) = make_float4(v[0], v[1], v[2], v[3]);
  }
}

__global__ __launch_bounds__(TPB) void roi_pool_kernel(
    const float* __restrict__ images,
    const float* __restrict__ rois,
    const int*   __restrict__ roi_idx,
    float* __restrict__ out,
    int n_rois)
{
  __shared__ __align__(16) float tbuf[2][TILE_F];   // ~57 KB, double buffered
  __shared__ int s_win[49];
  __shared__ int s_meta[5];                         // x1, y1, wspan, hspan, batch

  const int n   = blockIdx.x;
  const int tid = threadIdx.x;
  if (n >= n_rois) return;

  if (tid == 0) {
    const float4 r = ((const float4*)rois)[n];      // 16B-aligned ROI row
    int x1 = (int)floorf(r.x * (float)ROI_W);
    int y1 = (int)floorf(r.y * (float)ROI_H);
    int x2 = (int)ceilf (r.z * (float)ROI_W);
    int y2 = (int)ceilf (r.w * (float)ROI_H);
    s_meta[0] = x1; s_meta[1] = y1;
    s_meta[2] = x2 - x1; s_meta[3] = y2 - y1;
    s_meta[4] = roi_idx[n];
  }
  __syncthreads();

  const int x1 = s_meta[0], y1 = s_meta[1];
  const int wspan = s_meta[2], hspan = s_meta[3], b = s_meta[4];

  // Packed per-output-pixel window table: LDS offset | x-size-2 flag | y-size-2 flag.
  // Published to all waves by the barrier preceding the first compute below.
  if (tid < 49) {
    const int ph = tid / 7, pw = tid - ph * 7;
    const int ws = (pw * wspan) / ROI_P, we = ((pw + 1) * wspan + ROI_P - 1) / ROI_P;
    const int hs = (ph * hspan) / ROI_P, he = ((ph + 1) * hspan + ROI_P - 1) / ROI_P;
    int w = hs * ROI_W + x1 + ws;                   // rows stored from y1 -> hs is relative
    if (we - ws > 1) w |= 0x100;
    if (he - hs > 1) w |= 0x200;
    s_win[tid] = w;
  }

  const int spanq = hspan * (ROI_W / 2);            // u64 words per channel crop (21..56)
  const int cth   = tid >> 2;                       // channel this thread copies
  const int t4    = tid & 3;                        // 4 threads cooperate per channel
  const float* gbase = images
      + (size_t)b * ROI_C * ROI_H * ROI_W
      + (size_t)cth * (ROI_H * ROI_W)
      + (size_t)y1 * ROI_W;                         // 8B-aligned
  float* outn = out + (size_t)n * ROI_C * 49;

#if HAVE_ASYNC_B64
  stage_chunk_async(gbase, &tbuf[0][cth * CH_F], spanq, t4);
  for (int chunk = 0; chunk < NCHUNK; ++chunk) {
    if (chunk + 1 < NCHUNK) {
      // prefetch next chunk into the other buffer, then wait only for the
      // previous batch (in-order async-load completion, <=14 newest pending)
      stage_chunk_async(gbase + (size_t)(chunk + 1) * CHUNK * (ROI_H * ROI_W),
                        &tbuf[(chunk + 1) & 1][cth * CH_F], spanq, t4);
      wait_async<14>();
    } else {
      wait_async<0>();
    }
    __syncthreads();                                 // all waves' loads for `chunk` done
    compute_chunk(tbuf[chunk & 1], s_win, outn + chunk * (CHUNK * 49));
    __syncthreads();                                 // readers done before buffer reuse
  }
#else
  for (int chunk = 0; chunk < NCHUNK; ++chunk) {
    const float* g = gbase + (size_t)chunk * CHUNK * (ROI_H * ROI_W);
    float* l = &tbuf[0][cth * CH_F];
    for (int r = t4; r < spanq; r += 4)
      ((float2*)l)[r] = ((const float2*)g)[r];
    __syncthreads();
    compute_chunk(tbuf[0], s_win, outn + chunk * (CHUNK * 49));
    __syncthreads();
  }
#endif
}

extern "C" void kernel_launch(void* const* d_in, const int* in_sizes, int n_in,
                              void* d_out, int out_size, void* d_ws, size_t ws_size,
                              hipStream_t stream) {
  (void)n_in; (void)out_size; (void)d_ws; (void)ws_size;
  const float* images  = (const float*)d_in[0];
  const float* rois    = (const float*)d_in[1];
  const int*   roi_idx = (const int*)d_in[2];
  float*       out     = (float*)d_out;
  const int N = in_sizes[2];                         // number of ROIs
  if (N <= 0) return;
  roi_pool_kernel<<<dim3((unsigned)N), dim3(TPB), 0, stream>>>(
      images, rois, roi_idx, out, N);
}